// MultiHeadNeighborhoodAttentionAggregator_6923487282405
// MI455X (gfx1250) — compile-verified
//
#include <hip/hip_runtime.h>
#include <hip/hip_bf16.h>
#include <math.h>
#include <stdint.h>

// ---------------------------------------------------------------------------
// MultiHeadNeighborhoodAttentionAggregator for MI455X (gfx1250, wave32)
//
// - logits[b,h,n] = (Wk_h^T @ q_h[b]) . neighbor[b,n]  (neighbor never projected)
// - V_WMMA_F32_16X16X4_F32 for all dense GEMMs, M-blocked 4x (64 rows/WG) so
//   each B-fragment (weights from L2) feeds 4 accumulators -> 4x less L2 traffic
// - TDM tensor_load_to_lds stages the center tile with LDS row padding
// ---------------------------------------------------------------------------

typedef __attribute__((ext_vector_type(2))) float    v2f;
typedef __attribute__((ext_vector_type(8))) float    v8f;
typedef __attribute__((ext_vector_type(4))) uint32_t u32x4;
typedef __attribute__((ext_vector_type(8))) uint32_t u32x8;

#define B_TOTAL 65536
#define KNB     16
#define DIM     256
#define HDIM    128
#define SCALEF  0.08838834764831845f   // 128^-0.5
#define NEGINF  (-1.0e9f)

#define MB      4                      // row-tiles (16 rows each) per WG
#define TB      (MB * 16)              // 64 batch rows per workgroup
#define NWAVES  8
#define NTHREADS (NWAVES * 32)

#define LDC     260                    // padded row stride (floats), 16B-multiple
#define LDG     516

// --- WMMA fragment helpers (ISA 7.12.2 layouts, wave32) --------------------
__device__ __forceinline__ v2f a_frag(const float* base, int ld, int k0, int lane) {
    int r  = lane & 15;
    int kh = (lane >> 4) << 1;
    const float* p = base + r * ld + k0 + kh;
    v2f a; a.x = p[0]; a.y = p[1];
    return a;
}
__device__ __forceinline__ v2f b_frag_rm(const float* base, int ld, int k0, int n0, int lane) {
    int n  = lane & 15;
    int kh = (lane >> 4) << 1;
    v2f b;
    b.x = base[(size_t)(k0 + kh    ) * ld + n0 + n];
    b.y = base[(size_t)(k0 + kh + 1) * ld + n0 + n];
    return b;
}
__device__ __forceinline__ v2f b_frag_tr(const float* base, int ld, int k0, int n0, int lane) {
    int n  = lane & 15;
    int kh = (lane >> 4) << 1;
    const float* p = base + (size_t)(n0 + n) * ld + k0 + kh;
    v2f b; b.x = p[0]; b.y = p[1];
    return b;
}
__device__ __forceinline__ void d_store_lds(float* base, int ld, int n0, int lane, v8f d) {
    int rbase = (lane >> 4) << 3;
    int n = n0 + (lane & 15);
#pragma unroll
    for (int i = 0; i < 8; ++i) base[(i + rbase) * ld + n] = d[i];
}
__device__ __forceinline__ v8f wmma4(v2f a, v2f b, v8f c) {
    return __builtin_amdgcn_wmma_f32_16x16x4_f32(false, a, false, b, (short)0, c,
                                                 false, false);
}

__global__ __launch_bounds__(NTHREADS)
void mhna_kernel(const float* __restrict__ center,      // [B,256]
                 const float* __restrict__ neighbor,    // [B,16,256]
                 const float* __restrict__ nweights,    // [B,16]
                 const int*   __restrict__ nmask,       // [B,16]
                 const float* __restrict__ Wq,          // [2,128,256]
                 const float* __restrict__ Wk,          // [2,128,256]
                 const float* __restrict__ Wout,        // [256,512]
                 const float* __restrict__ Wgate,       // [256,512]
                 const float* __restrict__ Wgateb,      // [256]
                 float* __restrict__ out)               // [B,256]
{
    __shared__ __align__(16) float sC [TB][LDC];   // center tile (whole kernel)
    __shared__ __align__(16) float sQP[TB][LDC];   // Q (P1/P2) then proj (P4/P5)
    __shared__ __align__(16) float sG [TB][LDG];   // qk (P2/P3) then multi_ctx
    __shared__ float sLC[TB][KNB];                 // masked log-confidence

    const int tid  = threadIdx.x;
    const int lane = tid & 31;
    const int wave = tid >> 5;
    const int r0   = blockIdx.x * TB;

    // ---- P0: TDM loads center tile (with LDS row padding); other waves
    //          compute masked log-confidence in parallel ---------------------
    if (wave == 0) {
        // Tensor DMA descriptor (ISA cdna5 ch.8): 1D tile of 64*256 fp32,
        // pad 4 DWORDs every 256 DWORDs -> reproduces sC[64][260] layout.
        uint64_t ga  = (uint64_t)(uintptr_t)(center + (size_t)r0 * DIM);
        uint32_t lds = (uint32_t)(uintptr_t)(&sC[0][0]);   // low 32b = LDS offset
        u32x4 g0;
        g0[0] = 1u;                                        // count=1 (valid D#)
        g0[1] = lds;                                       // lds_addr
        g0[2] = (uint32_t)ga;                              // global_addr[31:0]
        g0[3] = (uint32_t)((ga >> 32) & 0x01FFFFFFu) | 0x80000000u; // ga[56:32] | type=2
        u32x8 g1;
        g1[0] = (2u << 16)            // data_size = 4B
              | (1u << 20)            // pad_enable
              | (7u << 22)            // pad_interval = 256 DWORDs
              | (3u << 25);           // pad_amount   = 4 DWORDs
        g1[1] = (uint32_t)(TB * DIM) << 16;   // tensor_dim0[15:0] = 16384
        g1[2] = ((uint32_t)(TB * DIM) >> 16) | (1u << 16); // dim0 hi | tensor_dim1=1
        g1[3] = (uint32_t)(TB * DIM) << 16;   // tile_dim0 = 16384
        g1[4] = 1u;                           // tile_dim1 = 1
        g1[5] = (uint32_t)(TB * DIM);         // tensor_dim0_stride lo
        g1[6] = 0u;
        g1[7] = 0u;
        asm volatile("tensor_load_to_lds %0, %1" :: "s"(g0), "s"(g1) : "memory");
        __builtin_amdgcn_s_wait_tensorcnt(0);
    }
    for (int i = tid; i < TB * KNB; i += NTHREADS) {
        int row = i >> 4, kn = i & 15;
        size_t idx = (size_t)(r0 + row) * KNB + kn;
        float w = nweights[idx];
        int   m = nmask[idx];
        sLC[row][kn] = m ? logf(fmaxf(w, 1e-8f)) : NEGINF;
    }
    __syncthreads();

    // ---- P1: Q[64,256] = C x Wq^T  (B loaded once, 4 M-tiles per step) ----
    for (int t = wave; t < DIM / 16; t += NWAVES) {
        int n0 = t * 16;
        v8f acc[MB];
#pragma unroll
        for (int m = 0; m < MB; ++m) acc[m] = (v8f){0.f,0.f,0.f,0.f,0.f,0.f,0.f,0.f};
        for (int k0 = 0; k0 < DIM; k0 += 4) {
            v2f b = b_frag_tr(Wq, DIM, k0, n0, lane);
#pragma unroll
            for (int m = 0; m < MB; ++m) {
                v2f a = a_frag(&sC[m * 16][0], LDC, k0, lane);
                acc[m] = wmma4(a, b, acc[m]);
            }
        }
#pragma unroll
        for (int m = 0; m < MB; ++m) d_store_lds(&sQP[m * 16][0], LDC, n0, lane, acc[m]);
    }
    __syncthreads();

    // ---- P2: G_h[64,256] = Q_h x Wk_h -------------------------------------
    for (int t = wave; t < 2 * (DIM / 16); t += NWAVES) {
        int h  = t >> 4;
        int n0 = (t & 15) * 16;
        const float* Bk = Wk + (size_t)h * HDIM * DIM;
        v8f acc[MB];
#pragma unroll
        for (int m = 0; m < MB; ++m) acc[m] = (v8f){0.f,0.f,0.f,0.f,0.f,0.f,0.f,0.f};
        for (int k0 = 0; k0 < HDIM; k0 += 4) {
            v2f b = b_frag_rm(Bk, DIM, k0, n0, lane);
#pragma unroll
            for (int m = 0; m < MB; ++m) {
                v2f a = a_frag(&sQP[m * 16][h * HDIM], LDC, k0, lane);
                acc[m] = wmma4(a, b, acc[m]);
            }
        }
#pragma unroll
        for (int m = 0; m < MB; ++m) d_store_lds(&sG[m * 16][h * DIM], LDG, n0, lane, acc[m]);
    }
    __syncthreads();

    // ---- P3: attention (stream neighbor rows, wave32 softmax) -------------
    for (int rr = wave; rr < TB; rr += NWAVES) {
        const float* Nb = neighbor + (size_t)(r0 + rr) * KNB * DIM;
        __builtin_prefetch(Nb + lane * 128, 0, 1);   // global_prefetch_b8

        int h  = lane >> 4;
        int kn = lane & 15;
        const float*  gp = &sG[rr][h * DIM];
        const float4* np = (const float4*)(Nb + kn * DIM);
        float dot = 0.f;
#pragma unroll 8
        for (int d4 = 0; d4 < DIM / 4; ++d4) {
            float4 nv = np[d4];
            dot = fmaf(nv.x, gp[4 * d4 + 0],
                  fmaf(nv.y, gp[4 * d4 + 1],
                  fmaf(nv.z, gp[4 * d4 + 2],
                  fmaf(nv.w, gp[4 * d4 + 3], dot))));
        }
        float lc    = sLC[rr][kn];
        bool  valid = lc > 0.5f * NEGINF;
        float logit = valid ? fmaf(dot, SCALEF, lc) : NEGINF;

        float mx = logit;
        for (int m = 8; m >= 1; m >>= 1) mx = fmaxf(mx, __shfl_xor(mx, m));
        float e = valid ? __expf(logit - mx) : 0.f;
        float s = e;
        for (int m = 8; m >= 1; m >>= 1) s += __shfl_xor(s, m);
        float attn = e / s;

        int dcol = lane & 15;
        int hbas = (lane >> 4) * DIM;
        for (int j = 0; j < 16; ++j) {
            int d = dcol + (j << 4);
            float acc2 = 0.f;
#pragma unroll
            for (int kk = 0; kk < 16; ++kk) {
                float a = __shfl(attn, (lane & 16) + kk);
                acc2 = fmaf(a, Nb[kk * DIM + d], acc2);
            }
            sG[rr][hbas + d] = acc2;   // multi_ctx layout [h*256 + d]
        }
    }
    __syncthreads();

    // ---- P4: proj[64,256] = multi_ctx[64,512] x Wout^T --------------------
    for (int t = wave; t < DIM / 16; t += NWAVES) {
        int n0 = t * 16;
        v8f acc[MB];
#pragma unroll
        for (int m = 0; m < MB; ++m) acc[m] = (v8f){0.f,0.f,0.f,0.f,0.f,0.f,0.f,0.f};
        for (int k0 = 0; k0 < 2 * DIM; k0 += 4) {
            v2f b = b_frag_tr(Wout, 2 * DIM, k0, n0, lane);
#pragma unroll
            for (int m = 0; m < MB; ++m) {
                v2f a = a_frag(&sG[m * 16][0], LDG, k0, lane);
                acc[m] = wmma4(a, b, acc[m]);
            }
        }
#pragma unroll
        for (int m = 0; m < MB; ++m) d_store_lds(&sQP[m * 16][0], LDC, n0, lane, acc[m]);
    }
    __syncthreads();

    // ---- P5: gate = sigmoid([C|proj] x Wgate^T + b); fused output ---------
    for (int t = wave; t < DIM / 16; t += NWAVES) {
        int n0 = t * 16;
        v8f acc[MB];
#pragma unroll
        for (int m = 0; m < MB; ++m) acc[m] = (v8f){0.f,0.f,0.f,0.f,0.f,0.f,0.f,0.f};
        for (int k0 = 0; k0 < DIM; k0 += 4) {             // first half: center
            v2f b = b_frag_tr(Wgate, 2 * DIM, k0, n0, lane);
#pragma unroll
            for (int m = 0; m < MB; ++m) {
                v2f a = a_frag(&sC[m * 16][0], LDC, k0, lane);
                acc[m] = wmma4(a, b, acc[m]);
            }
        }
        for (int k0 = DIM; k0 < 2 * DIM; k0 += 4) {       // second half: proj
            v2f b = b_frag_tr(Wgate, 2 * DIM, k0, n0, lane);
#pragma unroll
            for (int m = 0; m < MB; ++m) {
                v2f a = a_frag(&sQP[m * 16][0], LDC, k0 - DIM, lane);
                acc[m] = wmma4(a, b, acc[m]);
            }
        }
        int o     = n0 + (lane & 15);
        int rbase = (lane >> 4) << 3;
        float bias = Wgateb[o];
#pragma unroll
        for (int m = 0; m < MB; ++m) {
#pragma unroll
            for (int i = 0; i < 8; ++i) {
                int row = m * 16 + i + rbase;
                float x = acc[m][i] + bias;
                float g = 1.f / (1.f + __expf(-x));
                float c = sC[row][o];
                float p = sQP[row][o];
                out[(size_t)(r0 + row) * DIM + o] = fmaf(g, c - p, p);
            }
        }
    }
}

extern "C" void kernel_launch(void* const* d_in, const int* in_sizes, int n_in,
                              void* d_out, int out_size, void* d_ws, size_t ws_size,
                              hipStream_t stream) {
    (void)in_sizes; (void)n_in; (void)out_size; (void)d_ws; (void)ws_size;
    const float* center   = (const float*)d_in[0];
    const float* neighbor = (const float*)d_in[1];
    const float* nweights = (const float*)d_in[2];
    const int*   nmask    = (const int*)d_in[3];
    const float* Wq       = (const float*)d_in[4];
    const float* Wk       = (const float*)d_in[5];
    const float* Wout     = (const float*)d_in[6];
    const float* Wgate    = (const float*)d_in[7];
    const float* Wgateb   = (const float*)d_in[8];
    float* out = (float*)d_out;

    dim3 grid(B_TOTAL / TB);   // 1024 workgroups x 64 rows
    dim3 block(NTHREADS);      // 8 waves (wave32)
    mhna_kernel<<<grid, block, 0, stream>>>(center, neighbor, nweights, nmask,
                                            Wq, Wk, Wout, Wgate, Wgateb, out);
}